// HyperCD_21320217657961
// MI455X (gfx1250) — compile-verified
//
#include <hip/hip_runtime.h>
#include <math.h>

// ---------------------------------------------------------------------------
// HyperCD / NCD-style pipeline for MI455X (gfx1250, wave32).
// Graph convs: edge-parallel SPMM with L2-resident f32 atomics.
// Dense GEMMs: V_WMMA_F32_16X16X4_F32, one 16x64 C-strip (4 tiles) per wave
// so each A fragment feeds 4 WMMAs.
// ---------------------------------------------------------------------------

typedef __attribute__((ext_vector_type(2))) float v2f;
typedef __attribute__((ext_vector_type(8))) float v8f;

#define LEAKY_F 0.8f
#define RES_F   0.8f

// ---------------------------- elementwise helpers --------------------------

__global__ void scale_copy_kernel(float* __restrict__ dst,
                                  const float* __restrict__ src,
                                  float s, int n) {
  int i = blockIdx.x * blockDim.x + threadIdx.x;
  if (i < n) dst[i] = s * src[i];
}

__global__ void add_inplace_kernel(float* __restrict__ acc,
                                   const float* __restrict__ x, int n) {
  int i = blockIdx.x * blockDim.x + threadIdx.x;
  if (i < n) acc[i] += x[i];
}

__global__ void sub_inplace_kernel(float* __restrict__ a,
                                   const float* __restrict__ b, int n) {
  int i = blockIdx.x * blockDim.x + threadIdx.x;
  if (i < n) a[i] -= b[i];
}

// ------------------------------- SPMM (COO) --------------------------------
// y[row[e], :] += val[e] * x[col[e], :]   (DE = 64, 16 threads/edge x 4 dims)

__global__ void spmm_atomic_kernel(const int* __restrict__ row,
                                   const int* __restrict__ col,
                                   const float* __restrict__ val,
                                   const float* __restrict__ x,
                                   float* __restrict__ y, int nnz) {
  int t = blockIdx.x * blockDim.x + threadIdx.x;
  int e = t >> 4;
  if (e >= nnz) return;
  int d = (t & 15) << 2;
  int r = row[e];
  int c = col[e];
  float v = val[e];
  const float4 xv = *(const float4*)(x + (size_t)c * 64 + d);
  float* yp = y + (size_t)r * 64 + d;
  atomicAdd(yp + 0, v * xv.x);
  atomicAdd(yp + 1, v * xv.y);
  atomicAdd(yp + 2, v * xv.z);
  atomicAdd(yp + 3, v * xv.w);
}

// ------------------------- generic WMMA fp32 GEMM --------------------------
// C[M,N] = act( ascale * A[aidx[m], :K] @ B[K,N] + bias[N] )
// ACT: 0 = none, 1 = leaky_relu(0.8), 2 = tanh
// One wave computes a 16x64 strip (4 adjacent 16x16 tiles) so the per-lane
// A fragment is reused by 4 WMMAs per k-step.

template <int ACT>
__global__ void gemm_wmma_kernel(const float* __restrict__ A,
                                 const float* __restrict__ B,
                                 const float* __restrict__ bias,
                                 float* __restrict__ C,
                                 const int* __restrict__ aidx,
                                 float ascale, int M, int N, int K) {
  const int lane = threadIdx.x & 31;
  const int wave = threadIdx.x >> 5;
  const int stripsN = N >> 6;                      // strips of 64 columns
  const int strip = blockIdx.x * (blockDim.x >> 5) + wave;
  if (strip >= (M >> 4) * stripsN) return;
  const int tm = strip / stripsN;
  const int tn = strip - tm * stripsN;

  const int r = lane & 15;       // A: M within tile; B/C: N within sub-tile
  const int half = lane >> 4;    // half-wave split of K (A/B) and M (C)

  const int am = tm * 16 + r;
  const long arow = aidx ? (long)aidx[am] : (long)am;
  const float* Ap = A + (size_t)arow * K + 2 * half;  // A[m][k0+2*half .. +1]
  const int bn = tn * 64 + r;

  v8f acc[4] = {{}, {}, {}, {}};
  for (int k0 = 0; k0 < K; k0 += 4) {
    v2f a = *(const v2f*)(Ap + k0);
    a *= ascale;
    const float* B0 = B + (size_t)(k0 + 2 * half) * N + bn;
    const float* B1 = B0 + N;
#pragma unroll
    for (int c = 0; c < 4; ++c) {
      v2f b;
      b.x = B0[c * 16];
      b.y = B1[c * 16];
      acc[c] = __builtin_amdgcn_wmma_f32_16x16x4_f32(false, a, false, b,
                                                     (short)0, acc[c],
                                                     false, false);
    }
  }

  float bia[4];
#pragma unroll
  for (int c = 0; c < 4; ++c) bia[c] = bias[bn + c * 16];

#pragma unroll
  for (int j = 0; j < 8; ++j) {
    const int m = tm * 16 + j + 8 * half;
    float* Cr = C + (size_t)m * N + bn;
#pragma unroll
    for (int c = 0; c < 4; ++c) {
      float x = acc[c][j] + bia[c];
      if (ACT == 1) x = x >= 0.0f ? x : LEAKY_F * x;
      if (ACT == 2) x = tanhf(x);
      Cr[c * 16] = x;
    }
  }
}

// --------------------- state = disc * (diff @ kf^T) * q_table --------------
// S[b,n] = disc[b] * ( sum_k D[b,k] * KF[n,k] ) * qtab[qid[b]*NC + n]
// KF is row-major [NC x DF]; its transpose is the B-matrix, so B[k][n] is a
// contiguous float2 load from KF[n*DF + k]. 16x64 strip per wave.

__global__ void state_wmma_kernel(const float* __restrict__ D,
                                  const float* __restrict__ KF,
                                  const float* __restrict__ disc,
                                  const float* __restrict__ qtab,
                                  const int* __restrict__ qid,
                                  float* __restrict__ S,
                                  int Bn, int NC, int DF) {
  const int lane = threadIdx.x & 31;
  const int wave = threadIdx.x >> 5;
  const int stripsN = NC >> 6;
  const int strip = blockIdx.x * (blockDim.x >> 5) + wave;
  if (strip >= (Bn >> 4) * stripsN) return;
  const int tm = strip / stripsN;
  const int tn = strip - tm * stripsN;

  const int r = lane & 15;
  const int half = lane >> 4;

  const float* Ap = D + (size_t)(tm * 16 + r) * DF + 2 * half;
  // B[k][n] = KF[n][k]; 4 sub-tiles at n = tn*64 + c*16 + r
  const float* Bp = KF + (size_t)(tn * 64 + r) * DF + 2 * half;
  const size_t Bstep = (size_t)16 * DF;

  v8f acc[4] = {{}, {}, {}, {}};
  for (int k0 = 0; k0 < DF; k0 += 4) {
    v2f a = *(const v2f*)(Ap + k0);
#pragma unroll
    for (int c = 0; c < 4; ++c) {
      v2f b = *(const v2f*)(Bp + c * Bstep + k0);
      acc[c] = __builtin_amdgcn_wmma_f32_16x16x4_f32(false, a, false, b,
                                                     (short)0, acc[c],
                                                     false, false);
    }
  }

#pragma unroll
  for (int j = 0; j < 8; ++j) {
    const int m = tm * 16 + j + 8 * half;
    const float dm = disc[m];
    const float* qrow = qtab + (size_t)qid[m] * NC + tn * 64 + r;
    float* Sr = S + (size_t)m * NC + tn * 64 + r;
#pragma unroll
    for (int c = 0; c < 4; ++c) {
      Sr[c * 16] = dm * acc[c][j] * qrow[c * 16];
    }
  }
}

// -------------------------- N==1 vector dot kernels ------------------------

__global__ void disc_kernel(const float* __restrict__ ce_acc,
                            const int* __restrict__ qid,
                            const float* __restrict__ Wd,
                            const float* __restrict__ bd,
                            float* __restrict__ disc, int Bn, int K) {
  int t = blockIdx.x * blockDim.x + threadIdx.x;
  if (t >= Bn) return;
  const float* a = ce_acc + (size_t)qid[t] * K;
  float s = bd[0];
  for (int k = 0; k < K; ++k) s += 0.25f * a[k] * Wd[k];
  disc[t] = 1.0f / (1.0f + expf(-s));
}

__global__ void final_kernel(const float* __restrict__ H,
                             const float* __restrict__ W,
                             const float* __restrict__ b,
                             float* __restrict__ out, int Bn, int K) {
  int t = blockIdx.x * blockDim.x + threadIdx.x;
  if (t >= Bn) return;
  const float* h = H + (size_t)t * K;
  float s = b[0];
  for (int k = 0; k < K; ++k) s += h[k] * W[k];
  out[t] = 1.0f / (1.0f + expf(-s));
}

// ------------------------------- orchestration -----------------------------

extern "C" void kernel_launch(void* const* d_in, const int* in_sizes, int n_in,
                              void* d_out, int out_size, void* d_ws, size_t ws_size,
                              hipStream_t stream) {
  const int*   user_id     = (const int*)d_in[0];
  const int*   question_id = (const int*)d_in[1];
  const float* q_table     = (const float*)d_in[2];
  const int*   u_row = (const int*)d_in[3];
  const int*   u_col = (const int*)d_in[4];
  const float* u_val = (const float*)d_in[5];
  const int*   e_row = (const int*)d_in[6];
  const int*   e_col = (const int*)d_in[7];
  const float* e_val = (const float*)d_in[8];
  const int*   k_row = (const int*)d_in[9];
  const int*   k_col = (const int*)d_in[10];
  const float* k_val = (const float*)d_in[11];
  const float* student_emb   = (const float*)d_in[12];
  const float* exercise_emb  = (const float*)d_in[13];
  const float* knowledge_emb = (const float*)d_in[14];
  const float* W_s = (const float*)d_in[15];
  const float* b_s = (const float*)d_in[16];
  const float* W_e = (const float*)d_in[17];
  const float* b_e = (const float*)d_in[18];
  const float* W_k = (const float*)d_in[19];
  const float* b_k = (const float*)d_in[20];
  const float* W_d = (const float*)d_in[21];
  const float* b_d = (const float*)d_in[22];
  const float* M1_w = (const float*)d_in[23];
  const float* M1_b = (const float*)d_in[24];
  const float* M2_w = (const float*)d_in[25];
  const float* M2_b = (const float*)d_in[26];
  const float* M3_w = (const float*)d_in[27];
  const float* M3_b = (const float*)d_in[28];
  const float* M4_w = (const float*)d_in[29];
  const float* M4_b = (const float*)d_in[30];

  const int B  = in_sizes[0];
  const int DE = 64;
  const int NU = in_sizes[12] / DE;
  const int NQ = in_sizes[13] / DE;
  const int NC = in_sizes[14] / DE;
  const int DF = in_sizes[15] / DE;   // 128
  const int EU = in_sizes[3];
  const int EQ = in_sizes[6];
  const int EK = in_sizes[9];
  const int H1 = in_sizes[24];        // 512
  const int H2 = in_sizes[26];        // 256
  const int H3 = in_sizes[28];        // 128

  // --------- workspace carve (floats) ---------
  float* w = (float*)d_ws;
  auto take = [&](size_t n) { float* p = w; w += n; return p; };
  float* cs_a = take((size_t)NU * DE);
  float* cs_b = take((size_t)NU * DE);
  float* cs_c = take((size_t)NU * DE);
  float* ce_a = take((size_t)NQ * DE);
  float* ce_b = take((size_t)NQ * DE);
  float* ce_c = take((size_t)NQ * DE);
  float* ck_a = take((size_t)NC * DE);
  float* ck_b = take((size_t)NC * DE);
  float* ck_c = take((size_t)NC * DE);
  float* sf    = take((size_t)B * DF);   // becomes diff = sf - ef
  float* ef    = take((size_t)B * DF);
  float* kf    = take((size_t)NC * DF);
  float* discb = take((size_t)B);
  float* state = take((size_t)B * NC);
  float* h1    = take((size_t)B * H1);
  float* h2    = take((size_t)B * H2);
  float* h3    = take((size_t)B * H3);
  (void)ws_size; (void)n_in;

  const int TPB = 256;
  auto gridE = [&](int n) { return (n + TPB - 1) / TPB; };

  // --------- graph convolutions: acc = emb + sum_l all_e_l ---------
  auto conv = [&](const float* emb, const int* row, const int* col,
                  const float* val, int n, int nnz,
                  float* acc, float* bufA, float* bufB) {
    const int ne = n * DE;
    scale_copy_kernel<<<gridE(ne), TPB, 0, stream>>>(acc, emb, 1.0f, ne);
    scale_copy_kernel<<<gridE(ne), TPB, 0, stream>>>(bufA, emb, 1.0f, ne);
    float* cur = bufA;
    float* nxt = bufB;
    for (int l = 0; l < 3; ++l) {
      scale_copy_kernel<<<gridE(ne), TPB, 0, stream>>>(nxt, cur, RES_F, ne);
      spmm_atomic_kernel<<<gridE(nnz * 16), TPB, 0, stream>>>(row, col, val,
                                                              cur, nxt, nnz);
      add_inplace_kernel<<<gridE(ne), TPB, 0, stream>>>(acc, nxt, ne);
      float* t = cur; cur = nxt; nxt = t;
    }
  };
  conv(student_emb,   u_row, u_col, u_val, NU, EU, cs_a, cs_b, cs_c);
  conv(exercise_emb,  e_row, e_col, e_val, NQ, EQ, ce_a, ce_b, ce_c);
  conv(knowledge_emb, k_row, k_col, k_val, NC, EK, ck_a, ck_b, ck_c);

  // --------- WMMA GEMM launches (128 threads = 4 waves, 1 strip/wave) ------
  const int GTPB = 128;
  auto gridS = [&](int M, int N) {
    int strips = (M / 16) * (N / 64);
    return (strips + 3) / 4;
  };

  // sf = leaky( 0.25*cs_acc[user_id] @ W_s + b_s )   [B x DF]
  gemm_wmma_kernel<1><<<gridS(B, DF), GTPB, 0, stream>>>(
      cs_a, W_s, b_s, sf, user_id, 0.25f, B, DF, DE);
  // ef = leaky( 0.25*ce_acc[question_id] @ W_e + b_e )
  gemm_wmma_kernel<1><<<gridS(B, DF), GTPB, 0, stream>>>(
      ce_a, W_e, b_e, ef, question_id, 0.25f, B, DF, DE);
  // kf = leaky( 0.25*ck_acc @ W_k + b_k )            [NC x DF]
  gemm_wmma_kernel<1><<<gridS(NC, DF), GTPB, 0, stream>>>(
      ck_a, W_k, b_k, kf, nullptr, 0.25f, NC, DF, DE);
  // disc = sigmoid( 0.25*ce_acc[question_id] @ W_d + b_d )
  disc_kernel<<<gridE(B), TPB, 0, stream>>>(ce_a, question_id, W_d, b_d,
                                            discb, B, DE);
  // diff = sf - ef (in place in sf)
  sub_inplace_kernel<<<gridE(B * DF), TPB, 0, stream>>>(sf, ef, B * DF);
  // state = disc * (diff @ kf^T) * q_table[question_id]   [B x NC]
  state_wmma_kernel<<<gridS(B, NC), GTPB, 0, stream>>>(
      sf, kf, discb, q_table, question_id, state, B, NC, DF);
  // MLP
  gemm_wmma_kernel<2><<<gridS(B, H1), GTPB, 0, stream>>>(
      state, M1_w, M1_b, h1, nullptr, 1.0f, B, H1, NC);
  gemm_wmma_kernel<2><<<gridS(B, H2), GTPB, 0, stream>>>(
      h1, M2_w, M2_b, h2, nullptr, 1.0f, B, H2, H1);
  gemm_wmma_kernel<2><<<gridS(B, H3), GTPB, 0, stream>>>(
      h2, M3_w, M3_b, h3, nullptr, 1.0f, B, H3, H2);
  final_kernel<<<gridE(B), TPB, 0, stream>>>(h3, M4_w, M4_b, (float*)d_out,
                                             B, H3);
}